// INNLightGCNLinkPredictor_42064909697221
// MI455X (gfx1250) — compile-verified
//
#include <hip/hip_runtime.h>
#include <math.h>

#define N_ENT 100000
#define DIM   128
#define BPOS  4096
#define KNEG  64

typedef __attribute__((ext_vector_type(2))) float v2f;
typedef __attribute__((ext_vector_type(8))) float v8f;

__device__ __forceinline__ float softplusf(float x) {
    return (x > 20.0f) ? x : log1pf(expf(x));
}

// ---------------- elementwise kernels (float4 vectorized) ----------------

__global__ void zero_f4(float4* __restrict__ p, long n4) {
    long i = (long)blockIdx.x * blockDim.x + threadIdx.x;
    long stride = (long)gridDim.x * blockDim.x;
    float4 z = make_float4(0.f, 0.f, 0.f, 0.f);
    for (; i < n4; i += stride) p[i] = z;
}

__global__ void softplus_f4(const float4* __restrict__ in, float4* __restrict__ out, long n4) {
    long i = (long)blockIdx.x * blockDim.x + threadIdx.x;
    long stride = (long)gridDim.x * blockDim.x;
    for (; i < n4; i += stride) {
        float4 v = in[i];
        float4 o;
        o.x = softplusf(v.x); o.y = softplusf(v.y);
        o.z = softplusf(v.z); o.w = softplusf(v.w);
        out[i] = o;
    }
}

__global__ void interval_relu_f4(const float4* __restrict__ C, const float4* __restrict__ R,
                                 float4* __restrict__ c, float4* __restrict__ r, long n4) {
    long i = (long)blockIdx.x * blockDim.x + threadIdx.x;
    long stride = (long)gridDim.x * blockDim.x;
    for (; i < n4; i += stride) {
        float4 cc = C[i], rr = R[i];
        float4 oc, orr;
        {
            float lo = fmaxf(cc.x - rr.x, 0.f), hi = fmaxf(cc.x + rr.x, 0.f);
            oc.x = 0.5f * (hi + lo); orr.x = 0.5f * (hi - lo);
        }
        {
            float lo = fmaxf(cc.y - rr.y, 0.f), hi = fmaxf(cc.y + rr.y, 0.f);
            oc.y = 0.5f * (hi + lo); orr.y = 0.5f * (hi - lo);
        }
        {
            float lo = fmaxf(cc.z - rr.z, 0.f), hi = fmaxf(cc.z + rr.z, 0.f);
            oc.z = 0.5f * (hi + lo); orr.z = 0.5f * (hi - lo);
        }
        {
            float lo = fmaxf(cc.w - rr.w, 0.f), hi = fmaxf(cc.w + rr.w, 0.f);
            oc.w = 0.5f * (hi + lo); orr.w = 0.5f * (hi - lo);
        }
        c[i] = oc; r[i] = orr;
    }
}

// ---------------- dual GEMM via f32 WMMA ----------------
// Zc = Ac @ W^T ; Zr = Ar @ |W|^T        (M = N_ENT rows, K = N = DIM = 128)
// Block: 256 threads = 8 waves. Block handles a 16-row M strip; wave w handles
// N-tile [16w, 16w+16). K swept in steps of 4 with V_WMMA_F32_16X16X4_F32.
__global__ void __launch_bounds__(256)
gemm_dual_wmma(const float* __restrict__ Ac, const float* __restrict__ Ar,
               const float* __restrict__ W,
               float* __restrict__ Zc, float* __restrict__ Zr) {
    __shared__ float sA[2][16 * 130];   // padded stride 130 -> conflict-free b64 reads

    const int mtile = blockIdx.x * 16;
    const int wave  = threadIdx.x >> 5;     // 0..7
    const int lane  = threadIdx.x & 31;
    const int ntile = wave * 16;

    // cooperative stage of the 16x128 A strips (c and r) into LDS
    {
        const int row = threadIdx.x >> 4;          // 0..15
        const int col = (threadIdx.x & 15) * 8;    // 0..120
        const float* gc = Ac + (size_t)(mtile + row) * DIM + col;
        const float* gr = Ar + (size_t)(mtile + row) * DIM + col;
        float4 c0 = *(const float4*)(gc);
        float4 c1 = *(const float4*)(gc + 4);
        float4 r0 = *(const float4*)(gr);
        float4 r1 = *(const float4*)(gr + 4);
        float* dc = &sA[0][row * 130 + col];
        float* dr = &sA[1][row * 130 + col];
        *(float4*)(dc)     = c0; *(float4*)(dc + 4) = c1;
        *(float4*)(dr)     = r0; *(float4*)(dr + 4) = r1;
    }
    __syncthreads();

    const int mrow  = lane & 15;
    const int khalf = (lane >> 4) * 2;          // 0 or 2: K sub-pair per half-wave

    const float* lAc = &sA[0][mrow * 130];
    const float* lAr = &sA[1][mrow * 130];
    const float* bW  = W + (size_t)(ntile + mrow) * DIM;   // B[k][n] = W[n][k]

    v8f accc = {}; v8f accr = {};
    #pragma unroll 8
    for (int kk = 0; kk < DIM; kk += 4) {
        v2f a  = *(const v2f*)(lAc + kk + khalf);
        v2f ar = *(const v2f*)(lAr + kk + khalf);
        v2f b  = *(const v2f*)(bW  + kk + khalf);
        v2f ba;
        ba.x = __builtin_fabsf(b.x);
        ba.y = __builtin_fabsf(b.y);
        accc = __builtin_amdgcn_wmma_f32_16x16x4_f32(false, a,  false, b,  (short)0, accc, false, false);
        accr = __builtin_amdgcn_wmma_f32_16x16x4_f32(false, ar, false, ba, (short)0, accr, false, false);
    }
    __syncthreads();   // all reads of this block's A strip done before stores (enables in-place reuse patterns)

    // D layout: VGPR v -> row (v + 8*(lane>>4)), col = lane&15
    const int rbase = mtile + ((lane >> 4) << 3);
    const int col   = ntile + (lane & 15);
    #pragma unroll
    for (int v = 0; v < 8; ++v) {
        Zc[(size_t)(rbase + v) * DIM + col] = accc[v];
        Zr[(size_t)(rbase + v) * DIM + col] = accr[v];
    }
}

// ---------------- edge scatter-add (segment_sum) ----------------
// One wave per edge; lane covers 4 dims. Accumulators live in L2 (51.2MB each).
__global__ void __launch_bounds__(256)
scatter_edges(const int* __restrict__ erow, const int* __restrict__ ecol,
              const float* __restrict__ ew,
              const float* __restrict__ Zc, const float* __restrict__ Zr,
              float* __restrict__ C, float* __restrict__ R, int E) {
    int e = blockIdx.x * 8 + (threadIdx.x >> 5);
    if (e >= E) return;
    const int lane = threadIdx.x & 31;
    const int row = erow[e];
    const int col = ecol[e];
    const float w  = ew[e];
    const float wa = fabsf(w);
    const float4 zc = *(const float4*)(Zc + (size_t)col * DIM + lane * 4);
    const float4 zr = *(const float4*)(Zr + (size_t)col * DIM + lane * 4);
    float* cd = C + (size_t)row * DIM + lane * 4;
    float* rd = R + (size_t)row * DIM + lane * 4;
    atomicAdd(cd + 0, w * zc.x);  atomicAdd(cd + 1, w * zc.y);
    atomicAdd(cd + 2, w * zc.z);  atomicAdd(cd + 3, w * zc.w);
    atomicAdd(rd + 0, wa * zr.x); atomicAdd(rd + 1, wa * zr.y);
    atomicAdd(rd + 2, wa * zr.z); atomicAdd(rd + 3, wa * zr.w);
}

// ---------------- scoring ----------------
__device__ __forceinline__ float score_one(const float* __restrict__ uc, const float* __restrict__ ur,
                                           const float* __restrict__ relc, const float* __restrict__ relr,
                                           int h, int rel, int t, int lane) {
    const int d = lane * 4;
    const float4 hc = *(const float4*)(uc + (size_t)h * DIM + d);
    const float4 hr = *(const float4*)(ur + (size_t)h * DIM + d);
    const float4 tc = *(const float4*)(uc + (size_t)t * DIM + d);
    const float4 tr = *(const float4*)(ur + (size_t)t * DIM + d);
    const float4 rc = *(const float4*)(relc + (size_t)rel * DIM + d);
    const float4 ro = *(const float4*)(relr + (size_t)rel * DIM + d);
    float s = 0.f;
    s += fabsf(hr.x + softplusf(ro.x) + tr.x) - fabsf(hc.x + rc.x - tc.x);
    s += fabsf(hr.y + softplusf(ro.y) + tr.y) - fabsf(hc.y + rc.y - tc.y);
    s += fabsf(hr.z + softplusf(ro.z) + tr.z) - fabsf(hc.z + rc.z - tc.z);
    s += fabsf(hr.w + softplusf(ro.w) + tr.w) - fabsf(hc.w + rc.w - tc.w);
    #pragma unroll
    for (int off = 16; off > 0; off >>= 1) s += __shfl_down(s, off, 32);
    return s;
}

__global__ void __launch_bounds__(256)
score_pos(const int* __restrict__ pos, const float* __restrict__ uc, const float* __restrict__ ur,
          const float* __restrict__ relc, const float* __restrict__ relr, float* __restrict__ out) {
    const int b = blockIdx.x * 8 + (threadIdx.x >> 5);   // BPOS divisible by 8
    const int lane = threadIdx.x & 31;
    const int h   = pos[b * 3 + 0];
    const int rel = pos[b * 3 + 1];
    const int t   = pos[b * 3 + 2];
    float s = score_one(uc, ur, relc, relr, h, rel, t, lane);
    if (lane == 0) out[b] = s;
}

__global__ void __launch_bounds__(256)
score_neg(const int* __restrict__ neg, const int* __restrict__ pos,
          const float* __restrict__ uc, const float* __restrict__ ur,
          const float* __restrict__ relc, const float* __restrict__ relr,
          float* __restrict__ out, int T) {
    const int idx = blockIdx.x * 8 + (threadIdx.x >> 5);
    if (idx >= T) return;
    const int lane = threadIdx.x & 31;
    const int h   = neg[(size_t)idx * 3 + 0];
    const int t   = neg[(size_t)idx * 3 + 2];
    const int rel = pos[(idx / KNEG) * 3 + 1];   // reference uses the POS relation
    float s = score_one(uc, ur, relc, relr, h, rel, t, lane);
    if (lane == 0) out[idx] = s;
}

// ---------------- orchestration ----------------
extern "C" void kernel_launch(void* const* d_in, const int* in_sizes, int n_in,
                              void* d_out, int out_size, void* d_ws, size_t ws_size,
                              hipStream_t stream) {
    const int*   pos    = (const int*)  d_in[0];
    const int*   neg    = (const int*)  d_in[1];
    const int*   erow   = (const int*)  d_in[2];
    const int*   ecol   = (const int*)  d_in[3];
    const float* ew     = (const float*)d_in[4];
    const float* ec     = (const float*)d_in[5];   // entity_center
    const float* erho   = (const float*)d_in[6];   // entity_rho
    const float* relc   = (const float*)d_in[7];
    const float* relrho = (const float*)d_in[8];
    const float* W1     = (const float*)d_in[9];
    const float* W2     = (const float*)d_in[10];
    float* out = (float*)d_out;

    const int E = in_sizes[2];               // 1,100,000 edges
    const int T = in_sizes[1] / 3;           // 262,144 negative triplets

    const size_t nelem = (size_t)N_ENT * DIM;      // 12.8M floats = 51.2 MB
    float* buf0 = (float*)d_ws;                    // 5 x 51.2 MB = 256 MB workspace
    float* buf1 = buf0 + nelem;
    float* buf2 = buf1 + nelem;
    float* buf3 = buf2 + nelem;
    float* buf4 = buf3 + nelem;

    const long n4 = (long)(nelem / 4);
    const dim3 blk(256);
    const dim3 gEw((unsigned)((n4 + 255) / 256));           // elementwise grid
    const dim3 gGemm(N_ENT / 16);                           // 6250 M-tiles
    const dim3 gScat((unsigned)((E + 7) / 8));
    const dim3 gPos(BPOS / 8);
    const dim3 gNeg((unsigned)((T + 7) / 8));

    // r0 = softplus(entity_rho)
    softplus_f4<<<gEw, blk, 0, stream>>>((const float4*)erho, (float4*)buf0, n4);

    // ----- layer 1 -----
    gemm_dual_wmma<<<gGemm, blk, 0, stream>>>(ec, buf0, W1, buf1, buf2);      // Zc,Zr
    zero_f4<<<gEw, blk, 0, stream>>>((float4*)buf3, 2 * n4);                  // C,R := 0 (contiguous)
    scatter_edges<<<gScat, blk, 0, stream>>>(erow, ecol, ew, buf1, buf2, buf3, buf4, E);
    interval_relu_f4<<<gEw, blk, 0, stream>>>((const float4*)buf3, (const float4*)buf4,
                                              (float4*)buf1, (float4*)buf2, n4);   // c1,r1

    // ----- layer 2 -----
    gemm_dual_wmma<<<gGemm, blk, 0, stream>>>(buf1, buf2, W2, buf3, buf4);    // Zc2,Zr2
    zero_f4<<<gEw, blk, 0, stream>>>((float4*)buf1, 2 * n4);                  // C2,R2 := 0
    scatter_edges<<<gScat, blk, 0, stream>>>(erow, ecol, ew, buf3, buf4, buf1, buf2, E);
    interval_relu_f4<<<gEw, blk, 0, stream>>>((const float4*)buf1, (const float4*)buf2,
                                              (float4*)buf3, (float4*)buf4, n4);   // u_c,u_r

    // ----- scoring -----
    score_pos<<<gPos, blk, 0, stream>>>(pos, buf3, buf4, relc, relrho, out);
    score_neg<<<gNeg, blk, 0, stream>>>(neg, pos, buf3, buf4, relc, relrho, out + BPOS, T);
}